// MDNLoss_4827543240953
// MI455X (gfx1250) — compile-verified
//
#include <hip/hip_runtime.h>
#include <hip/hip_bf16.h>
#include <math.h>
#include <stdint.h>

#define NEGV   (-10000.0f)
#define TSEQ   512
#define TMEL   1024
#define TT     16          // mel-time tile width
#define TSTRIDE 513        // 512 + 1 pad -> conflict-free column reads
#define NWAVES (TSEQ / 32)

// ---- CDNA5 async global->LDS path (compile-guarded; fallback = ds_store) ----
#if defined(__gfx1250__) && __has_builtin(__builtin_amdgcn_global_load_async_to_lds_b32)
#define USE_ASYNC_LDS 1
#endif

#define GLOBAL_AS __attribute__((address_space(1)))
#define LOCAL_AS  __attribute__((address_space(3)))
typedef GLOBAL_AS int* gintp;   // builtin expects global int* (per hipcc diagnostic)
typedef LOCAL_AS  int* lintp;   // and LDS int*

__device__ __forceinline__ float log_add_exp(float a, float b) {
    float m = fmaxf(a, b);
    float d = fabsf(a - b);
    return m + __logf(1.0f + __expf(-d));
}

__global__ __launch_bounds__(TSEQ) void mdn_forward_kernel(
    const float* __restrict__ logp,          // [B, TSEQ, TMEL]
    const int*   __restrict__ text_lengths,  // [B]
    const int*   __restrict__ mel_lengths,   // [B]
    float*       __restrict__ partial)       // [B] in d_ws
{
    __shared__ float tile[TT * TSTRIDE];     // tile[j][s]: column t0+j contiguous in s
    __shared__ float wedge[2][NWAVES];       // wave-edge alpha values (double buffered)

    const int b    = blockIdx.x;
    const int tid  = threadIdx.x;            // owns text position s == tid
    const int lane = tid & 31;
    const int wid  = tid >> 5;
    const int mel  = mel_lengths[b];         // uniform within block
    const int txt  = text_lengths[b];        // uniform within block

    const float* base = logp + (size_t)b * TSEQ * TMEL;

    float alpha = NEGV;
    int buf = 0;

    for (int t0 = 0; t0 < mel; t0 += TT) {
        __syncthreads();                     // previous tile fully consumed
        // Load tile: linear element e = tid + 512*i, j = e & 15, s = e >> 4.
        // Consecutive lanes -> consecutive j -> 64B-contiguous global chunks.
#pragma unroll
        for (int i = 0; i < TT; ++i) {
            int e = tid + TSEQ * i;
            int j = e & (TT - 1);
            int s = e >> 4;
            const float* g = base + (size_t)s * TMEL + (t0 + j);
            float* l = &tile[j * TSTRIDE + s];
#if USE_ASYNC_LDS
            // async DMA: global -> LDS, tracked by ASYNCcnt (CDNA5 path)
            __builtin_amdgcn_global_load_async_to_lds_b32(
                (gintp)(uintptr_t)g,
                (lintp)(uint32_t)(uintptr_t)l,
                0, 0);
#else
            *l = *g;
#endif
        }
#if USE_ASYNC_LDS
#if __has_builtin(__builtin_amdgcn_s_wait_asynccnt)
        __builtin_amdgcn_s_wait_asynccnt(0);
#else
        asm volatile("s_wait_asynccnt 0x0" ::: "memory");
#endif
#endif
        __syncthreads();                     // tile visible to all waves

        for (int j = 0; j < TT; ++j) {
            int t = t0 + j;
            if (t >= mel) break;             // uniform: all threads exit together
            float tv = tile[j * TSTRIDE + tid];
            if (t == 0) {
                alpha = (tid == 0) ? tv : NEGV;   // alpha0
            } else {
                // in-wave shift by one lane; cross-wave edge via LDS
                float up = __shfl_up(alpha, 1);
                if (lane == 31) wedge[buf][wid] = alpha;
                __syncthreads();             // single barrier per step (double-buffered)
                float left = (lane == 0)
                                 ? ((wid > 0) ? wedge[buf][wid - 1] : NEGV)
                                 : up;
                buf ^= 1;
                alpha = log_add_exp(alpha + 1e-4f, left + 1e-4f) + tv;
            }
            if (t == mel - 1 && tid == txt - 1)
                partial[b] = alpha;          // exactly one write per batch
        }
    }
}

__global__ __launch_bounds__(64) void mdn_reduce_kernel(
    const float* __restrict__ partial, float* __restrict__ out, int B)
{
    __shared__ float sh[64];
    int tid = threadIdx.x;
    sh[tid] = (tid < B) ? partial[tid] : 0.0f;
    __syncthreads();
#pragma unroll
    for (int off = 32; off > 0; off >>= 1) {
        if (tid < off) sh[tid] += sh[tid + off];
        __syncthreads();
    }
    if (tid == 0)
        out[0] = -sh[0] / ((float)B * (float)TSEQ);
}

extern "C" void kernel_launch(void* const* d_in, const int* in_sizes, int n_in,
                              void* d_out, int out_size, void* d_ws, size_t ws_size,
                              hipStream_t stream) {
    (void)n_in; (void)out_size; (void)ws_size;
    const float* logp  = (const float*)d_in[0];
    const int*   textl = (const int*)d_in[1];
    const int*   mell  = (const int*)d_in[2];
    const int    B     = in_sizes[1];        // text_lengths has B elements

    float* partial = (float*)d_ws;           // B floats of scratch
    mdn_forward_kernel<<<B, TSEQ, 0, stream>>>(logp, textl, mell, partial);
    mdn_reduce_kernel<<<1, 64, 0, stream>>>(partial, (float*)d_out, B);
}